// GeneralizedAdvantageEstimate_88691074662783
// MI455X (gfx1250) — compile-verified
//
#include <hip/hip_runtime.h>
#include <stdint.h>
#include <stddef.h>

// ---------------- Problem constants (match reference) ----------------
#define GAMMA_F 0.99f
#define LAMBD_F 0.95f

constexpr int T_DIM  = 2048;
constexpr int B_DIM  = 8192;
constexpr int CHUNKS = 32;                 // time chunks (affine-composed)
constexpr int CT     = T_DIM / CHUNKS;     // 64 timesteps per chunk
constexpr int BLOCK  = 256;                // 8 waves (wave32)
constexpr int VEC    = 4;                  // float4 per thread -> B128 ops
constexpr int COLS_PER_BLK = BLOCK * VEC;  // 1024 columns per workgroup
constexpr int NSTAGE = 4;                  // async LDS pipeline depth

typedef float v4f __attribute__((ext_vector_type(4)));
typedef int   vint4 __attribute__((vector_size(16)));   // matches builtin's V4i

#define AS1 __attribute__((address_space(1)))
#define AS3 __attribute__((address_space(3)))

// ---------------- CDNA5 async global->LDS copy (gfx1250) ----------------
// Builtin signature (probe-confirmed): (int4 AS1*, int4 AS3*, Imm, Imm)
__device__ __forceinline__ void async_load_b128(const void* gsrc, void* ldst) {
#if __has_builtin(__builtin_amdgcn_global_load_async_to_lds_b128)
  __builtin_amdgcn_global_load_async_to_lds_b128(
      (AS1 vint4*)(uintptr_t)gsrc,
      (AS3 vint4*)(uint32_t)(uintptr_t)ldst,
      0 /*imm offset*/, 0 /*cpol*/);
#else
  uint32_t l = (uint32_t)(uintptr_t)ldst;
  asm volatile("global_load_async_to_lds_b128 %0, %1, off"
               :: "v"(l), "v"(gsrc) : "memory");
#endif
}

__device__ __forceinline__ void wait_async_hi() { // allow NSTAGE-1 stages in flight
#if __has_builtin(__builtin_amdgcn_s_wait_asynccnt)
  __builtin_amdgcn_s_wait_asynccnt(3 * (NSTAGE - 1));
#else
  asm volatile("s_wait_asynccnt %0" :: "i"(3 * (NSTAGE - 1)));
#endif
}

__device__ __forceinline__ void wait_async_0() {
#if __has_builtin(__builtin_amdgcn_s_wait_asynccnt)
  __builtin_amdgcn_s_wait_asynccnt(0);
#else
  asm volatile("s_wait_asynccnt 0");
#endif
}

// ---------------- Chunked affine-scan kernel ----------------
// PASS1: compute per-chunk composition (D, C): adv_in -> D + C*adv_in
// PASS3: compose downstream partials into a carry, rescan chunk, write out.
template <bool PASS1>
__global__ __launch_bounds__(BLOCK)
void gae_scan(const float* __restrict__ rewards,
              const float* __restrict__ values,
              const float* __restrict__ masks,
              float* __restrict__ wsC,
              float* __restrict__ wsD,
              float* __restrict__ out)
{
  __shared__ v4f smem[NSTAGE][3][BLOCK];   // 4 stages x {r,m,v} x 4KB = 48KB

  const int tid  = threadIdx.x;
  const int col0 = (blockIdx.x * BLOCK + tid) * VEC;
  const int k    = blockIdx.y;             // chunk index
  const int tHi  = k * CT + (CT - 1);      // highest timestep in chunk

  // Per-thread base pointers at row tHi; row (tHi - i) = base - i*B_DIM.
  const size_t baseOff = (size_t)tHi * B_DIM + col0;
  const float* rBase = rewards + baseOff;
  const float* mBase = masks   + baseOff;
  const float* vBase = values  + baseOff;
  float*       oBase = out     + baseOff;

  auto issue = [&](int i, int stg) {       // stage for timestep t = tHi - i
    const size_t d = (size_t)i * B_DIM;
    async_load_b128(rBase - d, &smem[stg][0][tid]);
    async_load_b128(mBase - d, &smem[stg][1][tid]);
    async_load_b128(vBase - d, &smem[stg][2][tid]);
  };

#pragma unroll
  for (int i = 0; i < NSTAGE - 1; ++i) issue(i, i);   // pipeline prologue

  // V_{tHi+1} (values has T+1 rows, so this is always in range)
  v4f vnext = *(const v4f*)(vBase + B_DIM);

  v4f acc   = {0.f, 0.f, 0.f, 0.f};   // PASS1: D accumulator; PASS3: adv
  v4f cprod = {1.f, 1.f, 1.f, 1.f};   // PASS1: product of coefficients

  if constexpr (!PASS1) {
    // Fused "pass 2": compose partials of chunks k+1..CHUNKS-1 (L2-resident, 2MB)
    v4f carry = {0.f, 0.f, 0.f, 0.f};
    for (int j = CHUNKS - 1; j > k; --j) {
      const size_t poff = (size_t)j * B_DIM + col0;
      const v4f Dj = *(const v4f*)(wsD + poff);
      const v4f Cj = *(const v4f*)(wsC + poff);
      carry = Dj + Cj * carry;
    }
    acc = carry;                       // adv entering this chunk from above
  }

  static_assert(CT % NSTAGE == 0, "pipeline assumes CT multiple of NSTAGE");
  for (int base = 0; base < CT; base += NSTAGE) {
#pragma unroll
    for (int s = 0; s < NSTAGE; ++s) {        // fully unrolled: s is a literal
      const int i = base + s;
      if (i + (NSTAGE - 1) < CT) {
        issue(i + (NSTAGE - 1), (s + NSTAGE - 1) % NSTAGE);  // static stage idx
        wait_async_hi();
      } else {
        wait_async_0();                       // pipeline drain at chunk tail
      }
      __syncthreads();                        // stage-s data visible (all waves)

      const v4f r = smem[s][0][tid];          // ds_load_b128, static offsets
      const v4f m = smem[s][1][tid];
      const v4f v = smem[s][2][tid];
      __syncthreads();                        // safe to overwrite stage s next

      const v4f coef  = (GAMMA_F * LAMBD_F) * m;
      const v4f delta = r + GAMMA_F * m * vnext - v;
      acc = delta + coef * acc;               // backward affine recurrence
      if constexpr (PASS1) cprod *= coef;
      vnext = v;

      if constexpr (!PASS1) {
        // NT store: don't let the 64MB output evict inputs from the 192MB L2
        __builtin_nontemporal_store(acc, (v4f*)(oBase - (size_t)i * B_DIM));
      }
    }
  }

  if constexpr (PASS1) {
    const size_t poff = (size_t)k * B_DIM + col0;
    *(v4f*)(wsD + poff) = acc;
    *(v4f*)(wsC + poff) = cprod;
  }
}

// ---------------- Fallback (in case ws_size is tiny): plain column scan ----------------
__global__ void gae_fallback(const float* __restrict__ rewards,
                             const float* __restrict__ values,
                             const float* __restrict__ masks,
                             float* __restrict__ out)
{
  const int c0 = (blockIdx.x * blockDim.x + threadIdx.x) * VEC;
  if (c0 >= B_DIM) return;
  v4f adv   = {0.f, 0.f, 0.f, 0.f};
  v4f vnext = *(const v4f*)(values + (size_t)T_DIM * B_DIM + c0);
  for (int t = T_DIM - 1; t >= 0; --t) {
    const size_t off = (size_t)t * B_DIM + c0;
    const v4f r = *(const v4f*)(rewards + off);
    const v4f m = *(const v4f*)(masks + off);
    const v4f v = *(const v4f*)(values + off);
    const v4f delta = r + GAMMA_F * m * vnext - v;
    adv   = delta + (GAMMA_F * LAMBD_F) * m * adv;
    vnext = v;
    __builtin_nontemporal_store(adv, (v4f*)(out + off));
  }
}

// ---------------- Launch ----------------
extern "C" void kernel_launch(void* const* d_in, const int* in_sizes, int n_in,
                              void* d_out, int out_size, void* d_ws, size_t ws_size,
                              hipStream_t stream)
{
  const float* rewards = (const float*)d_in[0];   // [T, B]
  const float* values  = (const float*)d_in[1];   // [T+1, B]
  const float* masks   = (const float*)d_in[2];   // [T, B]
  float* out = (float*)d_out;                     // [T, B]

  const size_t wsNeed = (size_t)2 * CHUNKS * B_DIM * sizeof(float); // 2MB
  if (d_ws != nullptr && ws_size >= wsNeed) {
    float* wsC = (float*)d_ws;
    float* wsD = wsC + (size_t)CHUNKS * B_DIM;
    dim3 grid(B_DIM / COLS_PER_BLK, CHUNKS);      // 8 x 32 = 256 WGs, 2048 waves
    gae_scan<true ><<<grid, BLOCK, 0, stream>>>(rewards, values, masks, wsC, wsD, out);
    gae_scan<false><<<grid, BLOCK, 0, stream>>>(rewards, values, masks, wsC, wsD, out);
  } else {
    gae_fallback<<<(B_DIM / VEC + 255) / 256, 256, 0, stream>>>(rewards, values, masks, out);
  }
}